// Slot_Attention_89696097010164
// MI455X (gfx1250) — compile-verified
//
#include <hip/hip_runtime.h>
#include <hip/hip_bf16.h>
#include <math.h>

// ---- problem constants (from reference setup_inputs) ----
#define Bb   8
#define Nn   1024
#define Mm   4096
#define Cc   512
#define Kk   16
#define NEG_INF (-3.402823466e38f)

typedef __attribute__((ext_vector_type(2))) float v2f;
typedef __attribute__((ext_vector_type(8))) float v8f;

__device__ __forceinline__ v8f wmma_f32(v2f a, v2f b, v8f c) {
    // V_WMMA_F32_16X16X4_F32 : D = A(16x4,f32) * B(4x16,f32) + C(16x16,f32)
    return __builtin_amdgcn_wmma_f32_16x16x4_f32(
        /*neg_a=*/false, a, /*neg_b=*/false, b,
        /*c_mod=*/(short)0, c, /*reuse_a=*/false, /*reuse_b=*/false);
}

// ---------------------------------------------------------------------------
// Row L2-normalize: rows x 512, one wave per row (8 rows / 256-thread block)
// ---------------------------------------------------------------------------
__global__ void k_norm512(const float* __restrict__ in, float* __restrict__ out) {
    const int wave = threadIdx.x >> 5;
    const int lane = threadIdx.x & 31;
    const int row  = blockIdx.x * 8 + wave;
    const float* src = in + (size_t)row * Cc + lane * 16;
    float4 v[4];
    float ss = 0.f;
#pragma unroll
    for (int i = 0; i < 4; ++i) {
        v[i] = *(const float4*)(src + i * 4);
        ss += v[i].x*v[i].x + v[i].y*v[i].y + v[i].z*v[i].z + v[i].w*v[i].w;
    }
#pragma unroll
    for (int m = 1; m < 32; m <<= 1) ss += __shfl_xor(ss, m, 32);
    const float rn = rsqrtf(ss);
    float* dst = out + (size_t)row * Cc + lane * 16;
#pragma unroll
    for (int i = 0; i < 4; ++i) {
        float4 o; o.x = v[i].x*rn; o.y = v[i].y*rn; o.z = v[i].z*rn; o.w = v[i].w*rn;
        *(float4*)(dst + i * 4) = o;
    }
}

// ---------------------------------------------------------------------------
// GEMM: out(RxC) = A(RxC) @ W(CxC) [+ resid], C=512, fp32 WMMA 16x16x4.
// Each wave computes TWO 16x16 row tiles sharing one B operand pair
// (2 WMMAs per W load -> 2x arithmetic intensity on the B stream).
// grid = (R/32, 4), block = 256 (8 waves), wave w -> col tile blockIdx.y*8+w.
// ---------------------------------------------------------------------------
__global__ void k_gemm512(const float* __restrict__ A, const float* __restrict__ W,
                          const float* __restrict__ resid, float* __restrict__ out) {
    const int t    = threadIdx.x;
    const int wave = t >> 5;
    const int lane = t & 31;
    const int half = lane >> 4;      // 0: lanes 0-15, 1: lanes 16-31
    const int n16  = lane & 15;
    const int row0 = blockIdx.x * 32;
    const int col0 = (blockIdx.y * 8 + wave) * 16;

    v8f acc0 = {}, acc1 = {};
    if (resid) {
#pragma unroll
        for (int r = 0; r < 8; ++r) {
            acc0[r] = resid[(size_t)(row0 +      r + 8*half) * Cc + col0 + n16];
            acc1[r] = resid[(size_t)(row0 + 16 + r + 8*half) * Cc + col0 + n16];
        }
    }

    const float* Arow0 = A + (size_t)(row0 + n16) * Cc;        // tile 0, A row M=n16
    const float* Arow1 = A + (size_t)(row0 + 16 + n16) * Cc;   // tile 1
    const float* Wc    = W + col0 + n16;                       // B col N=n16
#pragma unroll 4
    for (int k0 = 0; k0 < Cc; k0 += 4) {
        const int kk = k0 + 2 * half;
        v2f b; b.x = Wc[(size_t)kk * Cc];                      // B[kk][n]
               b.y = Wc[(size_t)(kk + 1) * Cc];                // B[kk+1][n]
        v2f a0 = *(const v2f*)(Arow0 + kk);
        v2f a1 = *(const v2f*)(Arow1 + kk);
        acc0 = wmma_f32(a0, b, acc0);
        acc1 = wmma_f32(a1, b, acc1);
    }
#pragma unroll
    for (int r = 0; r < 8; ++r) {
        out[(size_t)(row0 +      r + 8*half) * Cc + col0 + n16] = acc0[r];
        out[(size_t)(row0 + 16 + r + 8*half) * Cc + col0 + n16] = acc1[r];
    }
}

// ---------------------------------------------------------------------------
// Fused normalize(slots tile) + corr = nt @ nr^T (WMMA) + running top-16.
// 32 slots per block: each nr (B-operand) load feeds TWO WMMAs, halving the
// dominant L2 stream (nr is read once per 32 slots instead of per 16).
// grid = (N/32, B), block = 128 (4 waves). Wave w sweeps m-tiles w, w+4, ...
// corr is never materialized in global memory.
// ---------------------------------------------------------------------------
__global__ void k_topk(const float* __restrict__ slots, const float* __restrict__ nr,
                       int* __restrict__ idxOut) {
    __shared__ float ntile[32][Cc];     // 64 KB  normalized slot tile (2 A tiles)
    __shared__ float scr[4][32][16];    //  8 KB  per-wave corr tiles [slot][refcol]
    __shared__ float topv[4][32][Kk];   //  8 KB  per-wave per-slot top values
    __shared__ int   topi[4][32][Kk];   //  8 KB  matching ref indices

    const int b     = blockIdx.y;
    const int slot0 = blockIdx.x * 32;
    const int t     = threadIdx.x;      // 0..127
    const int wave  = t >> 5;
    const int lane  = t & 31;

    // ---- stage 1: load 32 slot rows, L2-normalize into LDS ----
    {
        const int row = t >> 2;         // 0..31
        const int sub = t & 3;          // 0..3 (128 channels each)
        const float* src = slots + ((size_t)b * Nn + slot0 + row) * Cc + sub * 128;
        float ss = 0.f;
#pragma unroll
        for (int i = 0; i < 32; ++i) {
            float4 x = *(const float4*)(src + 4 * i);
            ss += x.x*x.x + x.y*x.y + x.z*x.z + x.w*x.w;
        }
        ss += __shfl_xor(ss, 1, 32);
        ss += __shfl_xor(ss, 2, 32);
        const float rn = rsqrtf(ss);
#pragma unroll
        for (int i = 0; i < 32; ++i) {
            float4 x = *(const float4*)(src + 4 * i);
            ntile[row][sub*128 + 4*i + 0] = x.x * rn;
            ntile[row][sub*128 + 4*i + 1] = x.y * rn;
            ntile[row][sub*128 + 4*i + 2] = x.z * rn;
            ntile[row][sub*128 + 4*i + 3] = x.w * rn;
        }
    }
    for (int j = t; j < 4 * 32 * Kk; j += 128) {
        (&topv[0][0][0])[j] = NEG_INF;
        (&topi[0][0][0])[j] = 0;
    }
    __syncthreads();

    const int half = lane >> 4;
    const int n16  = lane & 15;
    float curMin = NEG_INF;   // every lane owns slot s = lane (0..31)
    int   minPos = 0;

    for (int mt = wave; mt < Mm / 16; mt += 4) {
        const int m0 = mt * 16;
        const float* nrB = nr + ((size_t)b * Mm + m0 + n16) * Cc; // B col n = ref row
        if (mt + 4 < Mm / 16)   // cover L2 latency of next m-tile for this wave
            __builtin_prefetch(nrB + (size_t)64 * Cc, 0, 1);
        v8f acc0 = {}, acc1 = {};
#pragma unroll 4
        for (int k0 = 0; k0 < Cc; k0 += 4) {
            const int kk = k0 + 2 * half;
            v2f bv = *(const v2f*)(nrB + kk);
            v2f a0 = *(const v2f*)&ntile[n16][kk];
            v2f a1 = *(const v2f*)&ntile[16 + n16][kk];
            acc0 = wmma_f32(a0, bv, acc0);
            acc1 = wmma_f32(a1, bv, acc1);
        }
#pragma unroll
        for (int r = 0; r < 8; ++r) {
            scr[wave][     r + 8*half][n16] = acc0[r];   // slots 0..15
            scr[wave][16 + r + 8*half][n16] = acc1[r];   // slots 16..31
        }
        __syncthreads();

        // ---- running top-16, all 32 lanes (unsorted; softmax is perm-invariant)
        {
#pragma unroll
            for (int j = 0; j < 16; ++j) {
                const float v = scr[wave][lane][j];
                if (v > curMin) {
                    topv[wave][lane][minPos] = v;
                    topi[wave][lane][minPos] = m0 + j;
                    float mn = topv[wave][lane][0]; int mp = 0;
#pragma unroll
                    for (int p = 1; p < Kk; ++p) {
                        const float tv = topv[wave][lane][p];
                        if (tv < mn) { mn = tv; mp = p; }
                    }
                    curMin = mn; minPos = mp;
                }
            }
        }
        __syncthreads();
    }

    // ---- merge the 4 per-wave lists (wave 0; lane = slot, 32 slots) ----
    if (wave == 0) {
        for (int w = 1; w < 4; ++w) {
#pragma unroll
            for (int j = 0; j < Kk; ++j) {
                const float v = topv[w][lane][j];
                if (v > curMin) {
                    topv[0][lane][minPos] = v;
                    topi[0][lane][minPos] = topi[w][lane][j];
                    float mn = topv[0][lane][0]; int mp = 0;
#pragma unroll
                    for (int p = 1; p < Kk; ++p) {
                        const float tv = topv[0][lane][p];
                        if (tv < mn) { mn = tv; mp = p; }
                    }
                    curMin = mn; minPos = mp;
                }
            }
        }
        int* dst = idxOut + ((size_t)b * Nn + slot0 + lane) * Kk;
#pragma unroll
        for (int j = 0; j < Kk; ++j) dst[j] = topi[0][lane][j];
    }
}

// ---------------------------------------------------------------------------
// Per-slot attention over 16 gathered prototypes. One wave per slot row.
// o[row] = softmax(q[row]·refK[idx]/sqrt(C)) · refV[idx]
// ---------------------------------------------------------------------------
__global__ void k_attn(const float* __restrict__ q, const float* __restrict__ refK,
                       const float* __restrict__ refV, const int* __restrict__ idx,
                       float* __restrict__ o) {
    const int wave = threadIdx.x >> 5;
    const int lane = threadIdx.x & 31;
    const int row  = blockIdx.x * 8 + wave;     // 0 .. B*N-1
    const int b    = row >> 10;                 // N = 1024

    const float* qr = q + (size_t)row * Cc + lane * 16;
    float qs[16];
#pragma unroll
    for (int i = 0; i < 16; ++i) qs[i] = qr[i];

    const int* ip = idx + (size_t)row * Kk;
    int mi[Kk];
#pragma unroll
    for (int k = 0; k < Kk; ++k) mi[k] = ip[k];

    const float scale = 0.044194173824159216f;  // 1/sqrt(512)
    float logit[Kk];
#pragma unroll
    for (int k = 0; k < Kk; ++k) {
        const float* kr = refK + ((size_t)b * Mm + mi[k]) * Cc + lane * 16;
        float p = 0.f;
#pragma unroll
        for (int i = 0; i < 16; ++i) p += qs[i] * kr[i];
#pragma unroll
        for (int m = 1; m < 32; m <<= 1) p += __shfl_xor(p, m, 32);
        logit[k] = p * scale;                   // same value in all 32 lanes
    }
    float mx = logit[0];
#pragma unroll
    for (int k = 1; k < Kk; ++k) mx = fmaxf(mx, logit[k]);
    float den = 0.f;
#pragma unroll
    for (int k = 0; k < Kk; ++k) { logit[k] = __expf(logit[k] - mx); den += logit[k]; }
    const float rden = 1.f / den;

    float accv[16];
#pragma unroll
    for (int i = 0; i < 16; ++i) accv[i] = 0.f;
#pragma unroll
    for (int k = 0; k < Kk; ++k) {
        const float w = logit[k] * rden;
        const float* vr = refV + ((size_t)b * Mm + mi[k]) * Cc + lane * 16;
#pragma unroll
        for (int i = 0; i < 16; ++i) accv[i] += w * vr[i];
    }
    float* op = o + (size_t)row * Cc + lane * 16;
#pragma unroll
    for (int i = 0; i < 16; ++i) op[i] = accv[i];
}

// ---------------------------------------------------------------------------
extern "C" void kernel_launch(void* const* d_in, const int* in_sizes, int n_in,
                              void* d_out, int out_size, void* d_ws, size_t ws_size,
                              hipStream_t stream) {
    const float* slots_in = (const float*)d_in[0];
    const float* refList  = (const float*)d_in[1];
    const float* Wq = (const float*)d_in[2];
    const float* Wk = (const float*)d_in[3];
    const float* Wv = (const float*)d_in[4];
    const float* Wo = (const float*)d_in[5];
    float* out = (float*)d_out;

    const size_t BMC = (size_t)Bb * Mm * Cc;   // 16,777,216 floats
    const size_t BNC = (size_t)Bb * Nn * Cc;   //  4,194,304 floats
    const float* ref = refList + BMC;          // ref_pt_list[-1] (L=2)

    float* ws   = (float*)d_ws;                // ~252 MB of scratch used
    float* nr   = ws;                          // (B,M,C) normalized ref
    float* refK = nr + BMC;                    // (B,M,C) ref @ Wk (loop invariant!)
    float* refV = refK + BMC;                  // (B,M,C) ref @ Wv
    float* qb   = refV + BMC;                  // (B,N,C)
    float* ob   = qb + BNC;                    // (B,N,C) attention output
    float* s0   = ob + BNC;                    // (B,N,C) slots ping buffer
    int*   idxb = (int*)(s0 + BNC);            // (B,N,K)

    // ---- loop-invariant precompute (gather commutes with @Wk/@Wv) ----
    k_norm512<<<dim3((Bb*Mm)/8), 256, 0, stream>>>(ref, nr);
    k_gemm512<<<dim3((Bb*Mm)/32, 4), 256, 0, stream>>>(ref, Wk, nullptr, refK);
    k_gemm512<<<dim3((Bb*Mm)/32, 4), 256, 0, stream>>>(ref, Wv, nullptr, refV);

    // ---- 3 slot-attention iterations; final write lands in d_out ----
    const float* cur = slots_in;
    float* nxt[3] = { s0, out, out };          // it2 residual-GEMM is in-place safe
    for (int it = 0; it < 3; ++it) {
        k_topk<<<dim3(Nn/32, Bb), 128, 0, stream>>>(cur, nr, idxb);
        k_gemm512<<<dim3((Bb*Nn)/32, 4), 256, 0, stream>>>(cur, Wq, nullptr, qb);
        k_attn<<<dim3((Bb*Nn)/8), 256, 0, stream>>>(qb, refK, refV, idxb, ob);
        k_gemm512<<<dim3((Bb*Nn)/32, 4), 256, 0, stream>>>(ob, Wo, cur, nxt[it]);
        cur = nxt[it];
    }
}